// HierarchicalLatentModel_77730318123561
// MI455X (gfx1250) — compile-verified
//
#include <hip/hip_runtime.h>
#include <cmath>

// ---------------------------------------------------------------------------
// HierarchicalLatentModel on gfx1250 (MI455X), fp32 WMMA (16x16x4).
//
// Phase 1 (parallel over B*T rows): fast encoder + reparam + fast-GRU input
//   gates + state-part of slow encoder.   (all heavy GEMMs, WMMA; input tile
//   staged to LDS with GLOBAL_LOAD_ASYNC_TO_LDS_B128 when available)
// Phase 2 (sequential over T, 16 blocks x 1 wave, 16 batch rows each):
//   carry-dependent small GEMMs + GRU nonlinearities; recurrence weights
//   staged in LDS; per-step streaming operands prefetched into registers
//   ahead of the dependent WMMA chain.
// Phase 3 (parallel): decoder 48->256->256 from stored latents.
// ---------------------------------------------------------------------------

typedef float v2f __attribute__((ext_vector_type(2)));
typedef float v8f __attribute__((ext_vector_type(8)));

#define BB 256
#define TT 512
#define NR (BB * TT)  // 131072 rows

#if __has_builtin(__builtin_amdgcn_global_load_async_to_lds_b128) && \
    __has_builtin(__builtin_amdgcn_s_wait_asynccnt)
#define HAVE_ASYNC_LDS 1
typedef int v4i __attribute__((vector_size(4 * sizeof(int))));
typedef __attribute__((address_space(1))) v4i* gas_v4i_p;  // global
typedef __attribute__((address_space(3))) v4i* las_v4i_p;  // LDS
#endif

__device__ __forceinline__ v8f wmma4(v2f a, v2f b, v8f c) {
  // D = A(16x4, f32) x B(4x16, f32) + C(16x16, f32)
  return __builtin_amdgcn_wmma_f32_16x16x4_f32(false, a, false, b, (short)0, c,
                                               false, false);
}

__device__ __forceinline__ v8f vzero8() {
  v8f r;
  for (int i = 0; i < 8; ++i) r[i] = 0.f;
  return r;
}

__device__ __forceinline__ float sigmoid_f(float x) {
  return 1.f / (1.f + expf(-x));
}

__device__ __forceinline__ float gelu_f(float x) {
  // exact GELU: x * 0.5 * (1 + erf(x / sqrt(2)))
  return 0.5f * x * (1.f + erff(x * 0.70710678118654752f));
}

// ---------------------------------------------------------------------------
// Phase 1: per 16-row tile (row = b*T + t, contiguous).
//   s_state = X @ se_w1[:, :256]^T + se_b1                      -> ws
//   H  = gelu(X @ fe_w1^T + fe_b1)
//   fp = H @ fe_w2^T + fe_b2 ; z_fast_new = mean + exp(.5 lv)*nf
//   fgi = [z_fast_new, action] @ fg_wih^T + fg_bih              -> ws
// ---------------------------------------------------------------------------
__global__ __launch_bounds__(32) void phase1_kernel(
    const float* __restrict__ states, const float* __restrict__ actions,
    const float* __restrict__ noise_fast,
    const float* __restrict__ fe_w1, const float* __restrict__ fe_b1,
    const float* __restrict__ fe_w2, const float* __restrict__ fe_b2,
    const float* __restrict__ fg_wih, const float* __restrict__ fg_bih,
    const float* __restrict__ se_w1, const float* __restrict__ se_b1,
    float* __restrict__ fgi_ws, float* __restrict__ sstate_ws) {
  __shared__ float Xs[16 * 256];  // input tile, later reused as H tile
  __shared__ float Xf[16 * 36];   // [z_fast_new(32), action(4)]

  const int tid = threadIdx.x;
  const int lane16 = tid & 15;
  const int kh = (tid >> 4) & 1;  // lane half (K split for 16x16x4)
  const size_t r0 = (size_t)blockIdx.x * 16;

  // stage 16x256 input tile into LDS
#ifdef HAVE_ASYNC_LDS
  {
    const float4* src = (const float4*)(states + r0 * 256);
    float4* dst = (float4*)Xs;
    for (int i = tid; i < 16 * 64; i += 32) {
      __builtin_amdgcn_global_load_async_to_lds_b128(
          (gas_v4i_p)(src + i), (las_v4i_p)(dst + i), 0, 0);
    }
    __builtin_amdgcn_s_wait_asynccnt(0);
  }
#else
  {
    const float4* src = (const float4*)(states + r0 * 256);
    float4* dst = (float4*)Xs;
    for (int i = tid; i < 16 * 64; i += 32) dst[i] = src[i];
  }
#endif
  __syncthreads();

  // ---- s_state = X @ se_w1[:, :256]^T + se_b1  (128 cols, K=256) ----
  {
    v8f acc[8];
#pragma unroll
    for (int i = 0; i < 8; ++i) acc[i] = vzero8();
    for (int kk = 0; kk < 64; ++kk) {
      const int k = kk * 4 + 2 * kh;
      v2f a;
      a.x = Xs[lane16 * 256 + k];
      a.y = Xs[lane16 * 256 + k + 1];
#pragma unroll
      for (int nt = 0; nt < 8; ++nt) {
        const float* wp = se_w1 + (size_t)(nt * 16 + lane16) * 288 + k;
        v2f b;
        b.x = wp[0];
        b.y = wp[1];
        acc[nt] = wmma4(a, b, acc[nt]);
      }
    }
#pragma unroll
    for (int nt = 0; nt < 8; ++nt) {
      const int n = nt * 16 + lane16;
      const float bias = se_b1[n];
#pragma unroll
      for (int v = 0; v < 8; ++v) {
        const int m = v + 8 * kh;
        sstate_ws[(r0 + m) * 128 + n] = acc[nt][v] + bias;
      }
    }
  }

  // ---- H = gelu(X @ fe_w1^T + fe_b1)  (256 cols, K=256) ----
  {
    v8f acc[16];
#pragma unroll
    for (int i = 0; i < 16; ++i) acc[i] = vzero8();
    for (int kk = 0; kk < 64; ++kk) {
      const int k = kk * 4 + 2 * kh;
      v2f a;
      a.x = Xs[lane16 * 256 + k];
      a.y = Xs[lane16 * 256 + k + 1];
#pragma unroll
      for (int nt = 0; nt < 16; ++nt) {
        const float* wp = fe_w1 + (size_t)(nt * 16 + lane16) * 256 + k;
        v2f b;
        b.x = wp[0];
        b.y = wp[1];
        acc[nt] = wmma4(a, b, acc[nt]);
      }
    }
    __syncthreads();  // all lanes finished reading X before overwrite
#pragma unroll
    for (int nt = 0; nt < 16; ++nt) {
      const int n = nt * 16 + lane16;
      const float bias = fe_b1[n];
#pragma unroll
      for (int v = 0; v < 8; ++v) {
        const int m = v + 8 * kh;
        Xs[m * 256 + n] = gelu_f(acc[nt][v] + bias);
      }
    }
  }
  __syncthreads();

  // ---- fp = H @ fe_w2^T + fe_b2 ; z_fast_new = mean + exp(.5 lv) * nf ----
  {
    v8f acc[4];
#pragma unroll
    for (int i = 0; i < 4; ++i) acc[i] = vzero8();
    for (int kk = 0; kk < 64; ++kk) {
      const int k = kk * 4 + 2 * kh;
      v2f a;
      a.x = Xs[lane16 * 256 + k];
      a.y = Xs[lane16 * 256 + k + 1];
#pragma unroll
      for (int nt = 0; nt < 4; ++nt) {
        const float* wp = fe_w2 + (size_t)(nt * 16 + lane16) * 256 + k;
        v2f b;
        b.x = wp[0];
        b.y = wp[1];
        acc[nt] = wmma4(a, b, acc[nt]);
      }
    }
#pragma unroll
    for (int nt = 0; nt < 2; ++nt) {  // tiles 0,1 = mean ; 2,3 = logvar
      const int n = nt * 16 + lane16;
      const float bm = fe_b2[n];
      const float bl = fe_b2[32 + n];
#pragma unroll
      for (int v = 0; v < 8; ++v) {
        const int m = v + 8 * kh;
        const size_t rr = r0 + m;
        const float mean = acc[nt][v] + bm;
        const float lv = acc[nt + 2][v] + bl;
        Xf[m * 36 + n] = mean + expf(0.5f * lv) * noise_fast[rr * 32 + n];
      }
    }
    for (int i = tid; i < 16 * 4; i += 32) {
      const int m = i >> 2, c = i & 3;
      Xf[m * 36 + 32 + c] = actions[(r0 + m) * 4 + c];
    }
  }
  __syncthreads();

  // ---- fgi = [z_fast_new, action] @ fg_wih^T + fg_bih  (96 cols, K=36) ----
  {
    v8f acc[6];
#pragma unroll
    for (int i = 0; i < 6; ++i) acc[i] = vzero8();
    for (int kk = 0; kk < 9; ++kk) {
      const int k = kk * 4 + 2 * kh;
      v2f a;
      a.x = Xf[lane16 * 36 + k];
      a.y = Xf[lane16 * 36 + k + 1];
#pragma unroll
      for (int nt = 0; nt < 6; ++nt) {
        const float* wp = fg_wih + (size_t)(nt * 16 + lane16) * 36 + k;
        v2f b;
        b.x = wp[0];
        b.y = wp[1];
        acc[nt] = wmma4(a, b, acc[nt]);
      }
    }
#pragma unroll
    for (int nt = 0; nt < 6; ++nt) {
      const int n = nt * 16 + lane16;
      const float bias = fg_bih[n];
#pragma unroll
      for (int v = 0; v < 8; ++v) {
        const int m = v + 8 * kh;
        fgi_ws[(r0 + m) * 96 + n] = acc[nt][v] + bias;
      }
    }
  }
}

// ---------------------------------------------------------------------------
// Phase 2: sequential scan over T. One wave per 16 batch rows (16 blocks).
// Recurrence weights live in LDS; per-step operands register-prefetched.
// ---------------------------------------------------------------------------
__global__ __launch_bounds__(32) void phase2_kernel(
    const float* __restrict__ noise_slow, const float* __restrict__ fg_whh,
    const float* __restrict__ fg_bhh, const float* __restrict__ se_w1,
    const float* __restrict__ se_w2, const float* __restrict__ se_b2,
    const float* __restrict__ sg_wih, const float* __restrict__ sg_whh,
    const float* __restrict__ sg_bih, const float* __restrict__ sg_bhh,
    const float* __restrict__ upd_ptr, const float* __restrict__ fgi_ws,
    const float* __restrict__ sstate_ws, float* __restrict__ out_zf,
    float* __restrict__ out_zs) {
  __shared__ float hs[16 * 32];       // z_fast carry
  __shared__ float ss[16 * 16];       // z_slow carry
  __shared__ float Gs[16 * 128];      // slow-encoder hidden
  __shared__ float X2[16 * 48];       // [z_slow_new(16), z_fast(32)]
  __shared__ float w_fgwhh[96 * 32];  // 12 KB, loop-invariant weights
  __shared__ float w_sgwih[48 * 48];  //  9 KB
  __shared__ float w_sgwhh[48 * 16];  //  3 KB

  const int tid = threadIdx.x;
  const int lane16 = tid & 15;
  const int kh = (tid >> 4) & 1;
  const size_t b0 = (size_t)blockIdx.x * 16;
  const float upd = 1.f / (1.f + expf(-upd_ptr[0]));

  for (int i = tid; i < 16 * 32; i += 32) hs[i] = 0.f;
  for (int i = tid; i < 16 * 16; i += 32) ss[i] = 0.f;
  // stage recurrence weights into LDS once
  for (int i = tid; i < 96 * 32; i += 32) w_fgwhh[i] = fg_whh[i];
  for (int i = tid; i < 48 * 48; i += 32) w_sgwih[i] = sg_wih[i];
  for (int i = tid; i < 48 * 16; i += 32) w_sgwhh[i] = sg_whh[i];
  __syncthreads();

  for (int t = 0; t < TT; ++t) {
    // ---- prefetch this step's fast-GRU input gates into registers ----
    float gir[2][8], giz[2][8], gin[2][8];
#pragma unroll
    for (int j = 0; j < 2; ++j) {
      const int n = j * 16 + lane16;
#pragma unroll
      for (int v = 0; v < 8; ++v) {
        const size_t rr = (b0 + v + 8 * kh) * TT + t;
        const float* gp = fgi_ws + rr * 96;  // already includes fg_bih
        gir[j][v] = gp[n];
        giz[j][v] = gp[32 + n];
        gin[j][v] = gp[64 + n];
      }
    }

    // ---- fast GRU: gh = h @ fg_whh^T (96 cols, K=32) ----
    v8f gh[6];
#pragma unroll
    for (int i = 0; i < 6; ++i) gh[i] = vzero8();
    for (int kk = 0; kk < 8; ++kk) {
      const int k = kk * 4 + 2 * kh;
      v2f a;
      a.x = hs[lane16 * 32 + k];
      a.y = hs[lane16 * 32 + k + 1];
#pragma unroll
      for (int nt = 0; nt < 6; ++nt) {
        const float* wp = &w_fgwhh[(nt * 16 + lane16) * 32 + k];
        v2f b;
        b.x = wp[0];
        b.y = wp[1];
        gh[nt] = wmma4(a, b, gh[nt]);
      }
    }
    __syncthreads();  // all reads of hs complete before in-place update
#pragma unroll
    for (int j = 0; j < 2; ++j) {  // column tiles of the 32-wide hidden
      const int n = j * 16 + lane16;
      const float bhr = fg_bhh[n], bhz = fg_bhh[32 + n], bhn = fg_bhh[64 + n];
#pragma unroll
      for (int v = 0; v < 8; ++v) {
        const int m = v + 8 * kh;
        const size_t rr = (b0 + m) * TT + t;
        const float rg = sigmoid_f(gir[j][v] + gh[j][v] + bhr);
        const float zg = sigmoid_f(giz[j][v] + gh[j + 2][v] + bhz);
        const float ng = tanhf(gin[j][v] + rg * (gh[j + 4][v] + bhn));
        const float hold = hs[m * 32 + n];
        const float hv = (1.f - zg) * ng + zg * hold;
        hs[m * 32 + n] = hv;
        out_zf[rr * 32 + n] = hv;
      }
    }
    __syncthreads();

    // ---- prefetch s_state rows (independent of h) ----
    float sst[8][8];
#pragma unroll
    for (int nt = 0; nt < 8; ++nt) {
      const int n = nt * 16 + lane16;
#pragma unroll
      for (int v = 0; v < 8; ++v) {
        const size_t rr = (b0 + v + 8 * kh) * TT + t;
        sst[nt][v] = sstate_ws[rr * 128 + n];
      }
    }

    // ---- slow encoder: G = gelu(s_state + h @ se_w1[:,256:]^T) ----
    {
      v8f acc[8];
#pragma unroll
      for (int i = 0; i < 8; ++i) acc[i] = vzero8();
      for (int kk = 0; kk < 8; ++kk) {
        const int k = kk * 4 + 2 * kh;
        v2f a;
        a.x = hs[lane16 * 32 + k];
        a.y = hs[lane16 * 32 + k + 1];
#pragma unroll
        for (int nt = 0; nt < 8; ++nt) {
          const float* wp = se_w1 + (size_t)(nt * 16 + lane16) * 288 + 256 + k;
          v2f b;
          b.x = wp[0];
          b.y = wp[1];
          acc[nt] = wmma4(a, b, acc[nt]);
        }
      }
#pragma unroll
      for (int nt = 0; nt < 8; ++nt) {
        const int n = nt * 16 + lane16;
#pragma unroll
        for (int v = 0; v < 8; ++v) {
          const int m = v + 8 * kh;
          Gs[m * 128 + n] = gelu_f(acc[nt][v] + sst[nt][v]);
        }
      }
    }
    __syncthreads();

    // ---- sp = G @ se_w2^T + se_b2 (32 cols, K=128); reparam slow ----
    {
      v8f acc[2];
#pragma unroll
      for (int i = 0; i < 2; ++i) acc[i] = vzero8();
      for (int kk = 0; kk < 32; ++kk) {
        const int k = kk * 4 + 2 * kh;
        v2f a;
        a.x = Gs[lane16 * 128 + k];
        a.y = Gs[lane16 * 128 + k + 1];
#pragma unroll
        for (int nt = 0; nt < 2; ++nt) {
          const float* wp = se_w2 + (size_t)(nt * 16 + lane16) * 128 + k;
          v2f b;
          b.x = wp[0];
          b.y = wp[1];
          acc[nt] = wmma4(a, b, acc[nt]);
        }
      }
      const int n = lane16;
      const float bm = se_b2[n], bl = se_b2[16 + n];
#pragma unroll
      for (int v = 0; v < 8; ++v) {
        const int m = v + 8 * kh;
        const size_t rr = (b0 + m) * TT + t;
        const float mean = acc[0][v] + bm;
        const float lv = acc[1][v] + bl;
        X2[m * 48 + n] = mean + expf(0.5f * lv) * noise_slow[rr * 16 + n];
      }
      for (int i = tid; i < 16 * 32; i += 32) {
        const int m = i >> 5, c = i & 31;
        X2[m * 48 + 16 + c] = hs[m * 32 + c];
      }
    }
    __syncthreads();

    // ---- slow GRU (48 cols, K=48 and K=16) + blended update ----
    {
      v8f gi2[3], gh2[3];
#pragma unroll
      for (int i = 0; i < 3; ++i) {
        gi2[i] = vzero8();
        gh2[i] = vzero8();
      }
      for (int kk = 0; kk < 12; ++kk) {
        const int k = kk * 4 + 2 * kh;
        v2f a;
        a.x = X2[lane16 * 48 + k];
        a.y = X2[lane16 * 48 + k + 1];
#pragma unroll
        for (int nt = 0; nt < 3; ++nt) {
          const float* wp = &w_sgwih[(nt * 16 + lane16) * 48 + k];
          v2f b;
          b.x = wp[0];
          b.y = wp[1];
          gi2[nt] = wmma4(a, b, gi2[nt]);
        }
      }
      for (int kk = 0; kk < 4; ++kk) {
        const int k = kk * 4 + 2 * kh;
        v2f a;
        a.x = ss[lane16 * 16 + k];
        a.y = ss[lane16 * 16 + k + 1];
#pragma unroll
        for (int nt = 0; nt < 3; ++nt) {
          const float* wp = &w_sgwhh[(nt * 16 + lane16) * 16 + k];
          v2f b;
          b.x = wp[0];
          b.y = wp[1];
          gh2[nt] = wmma4(a, b, gh2[nt]);
        }
      }
      __syncthreads();  // reads of ss complete before in-place update
      const int n = lane16;
      const float bir = sg_bih[n], biz = sg_bih[16 + n], bin = sg_bih[32 + n];
      const float bhr = sg_bhh[n], bhz = sg_bhh[16 + n], bhn = sg_bhh[32 + n];
#pragma unroll
      for (int v = 0; v < 8; ++v) {
        const int m = v + 8 * kh;
        const size_t rr = (b0 + m) * TT + t;
        const float rg = sigmoid_f(gi2[0][v] + bir + gh2[0][v] + bhr);
        const float zg = sigmoid_f(gi2[1][v] + biz + gh2[1][v] + bhz);
        const float ng = tanhf(gi2[2][v] + bin + rg * (gh2[2][v] + bhn));
        const float sold = ss[m * 16 + n];
        const float strans = (1.f - zg) * ng + zg * sold;
        const float snew = upd * strans + (1.f - upd) * sold;
        ss[m * 16 + n] = snew;
        out_zs[rr * 16 + n] = snew;
      }
    }
    __syncthreads();
  }
}

// ---------------------------------------------------------------------------
// Phase 3: decoder pred = gelu([zf, zs] @ de_w1^T + b1) @ de_w2^T + b2
// ---------------------------------------------------------------------------
__global__ __launch_bounds__(32) void phase3_kernel(
    const float* __restrict__ zf, const float* __restrict__ zs,
    const float* __restrict__ de_w1, const float* __restrict__ de_b1,
    const float* __restrict__ de_w2, const float* __restrict__ de_b2,
    float* __restrict__ preds) {
  __shared__ float Zs[16 * 48];
  __shared__ float Hs[16 * 256];
  const int tid = threadIdx.x;
  const int lane16 = tid & 15;
  const int kh = (tid >> 4) & 1;
  const size_t r0 = (size_t)blockIdx.x * 16;

  for (int i = tid; i < 16 * 32; i += 32) {
    const int m = i >> 5, c = i & 31;
    Zs[m * 48 + c] = zf[(r0 + m) * 32 + c];
  }
  for (int i = tid; i < 16 * 16; i += 32) {
    const int m = i >> 4, c = i & 15;
    Zs[m * 48 + 32 + c] = zs[(r0 + m) * 16 + c];
  }
  __syncthreads();

  v8f acc[16];
#pragma unroll
  for (int i = 0; i < 16; ++i) acc[i] = vzero8();
  for (int kk = 0; kk < 12; ++kk) {  // K=48
    const int k = kk * 4 + 2 * kh;
    v2f a;
    a.x = Zs[lane16 * 48 + k];
    a.y = Zs[lane16 * 48 + k + 1];
#pragma unroll
    for (int nt = 0; nt < 16; ++nt) {
      const float* wp = de_w1 + (size_t)(nt * 16 + lane16) * 48 + k;
      v2f b;
      b.x = wp[0];
      b.y = wp[1];
      acc[nt] = wmma4(a, b, acc[nt]);
    }
  }
#pragma unroll
  for (int nt = 0; nt < 16; ++nt) {
    const int n = nt * 16 + lane16;
    const float bias = de_b1[n];
#pragma unroll
    for (int v = 0; v < 8; ++v) {
      const int m = v + 8 * kh;
      Hs[m * 256 + n] = gelu_f(acc[nt][v] + bias);
    }
  }
  __syncthreads();

#pragma unroll
  for (int i = 0; i < 16; ++i) acc[i] = vzero8();
  for (int kk = 0; kk < 64; ++kk) {  // K=256
    const int k = kk * 4 + 2 * kh;
    v2f a;
    a.x = Hs[lane16 * 256 + k];
    a.y = Hs[lane16 * 256 + k + 1];
#pragma unroll
    for (int nt = 0; nt < 16; ++nt) {
      const float* wp = de_w2 + (size_t)(nt * 16 + lane16) * 256 + k;
      v2f b;
      b.x = wp[0];
      b.y = wp[1];
      acc[nt] = wmma4(a, b, acc[nt]);
    }
  }
#pragma unroll
  for (int nt = 0; nt < 16; ++nt) {
    const int n = nt * 16 + lane16;
    const float bias = de_b2[n];
#pragma unroll
    for (int v = 0; v < 8; ++v) {
      const int m = v + 8 * kh;
      preds[(r0 + m) * 256 + n] = acc[nt][v] + bias;
    }
  }
}

// ---------------------------------------------------------------------------
extern "C" void kernel_launch(void* const* d_in, const int* in_sizes, int n_in,
                              void* d_out, int out_size, void* d_ws,
                              size_t ws_size, hipStream_t stream) {
  (void)in_sizes;
  (void)n_in;
  (void)out_size;
  (void)ws_size;

  const float* states = (const float*)d_in[0];
  const float* actions = (const float*)d_in[1];
  const float* noise_fast = (const float*)d_in[2];
  const float* noise_slow = (const float*)d_in[3];
  const float* fe_w1 = (const float*)d_in[4];
  const float* fe_b1 = (const float*)d_in[5];
  const float* fe_w2 = (const float*)d_in[6];
  const float* fe_b2 = (const float*)d_in[7];
  const float* fg_wih = (const float*)d_in[8];
  const float* fg_whh = (const float*)d_in[9];
  const float* fg_bih = (const float*)d_in[10];
  const float* fg_bhh = (const float*)d_in[11];
  const float* se_w1 = (const float*)d_in[12];
  const float* se_b1 = (const float*)d_in[13];
  const float* se_w2 = (const float*)d_in[14];
  const float* se_b2 = (const float*)d_in[15];
  const float* sg_wih = (const float*)d_in[16];
  const float* sg_whh = (const float*)d_in[17];
  const float* sg_bih = (const float*)d_in[18];
  const float* sg_bhh = (const float*)d_in[19];
  const float* de_w1 = (const float*)d_in[20];
  const float* de_b1 = (const float*)d_in[21];
  const float* de_w2 = (const float*)d_in[22];
  const float* de_b2 = (const float*)d_in[23];
  const float* upd = (const float*)d_in[24];

  float* out = (float*)d_out;
  float* out_preds = out;                                    // [B,T,256]
  float* out_zf = out + (size_t)NR * 256;                    // [B,T,32]
  float* out_zs = out + (size_t)NR * 256 + (size_t)NR * 32;  // [B,T,16]

  float* fgi_ws = (float*)d_ws;                 // [B*T, 96]
  float* sstate_ws = fgi_ws + (size_t)NR * 96;  // [B*T, 128]

  const int nblk = NR / 16;  // 8192 row-tiles

  phase1_kernel<<<nblk, 32, 0, stream>>>(
      states, actions, noise_fast, fe_w1, fe_b1, fe_w2, fe_b2, fg_wih, fg_bih,
      se_w1, se_b1, fgi_ws, sstate_ws);

  phase2_kernel<<<BB / 16, 32, 0, stream>>>(
      noise_slow, fg_whh, fg_bhh, se_w1, se_w2, se_b2, sg_wih, sg_whh, sg_bih,
      sg_bhh, upd, fgi_ws, sstate_ws, out_zf, out_zs);

  phase3_kernel<<<nblk, 32, 0, stream>>>(out_zf, out_zs, de_w1, de_b1, de_w2,
                                         de_b2, out_preds);
}